// QuanvolutionSelfAttentionClassifier_65481071398734
// MI455X (gfx1250) — compile-verified
//
#include <hip/hip_runtime.h>
#include <math.h>

typedef __attribute__((ext_vector_type(2))) float v2f;
typedef __attribute__((ext_vector_type(8))) float v8f;

#define FEAT_STRIDE 788  // 784 + 4 pad -> conflict-free 16-lane column reads for WMMA A tiles

__global__ __launch_bounds__(32)
void quanv_attn_cls_kernel(const float* __restrict__ x,
                           const float* __restrict__ conv_w,
                           const float* __restrict__ conv_b,
                           const float* __restrict__ lin_w,
                           const float* __restrict__ lin_b,
                           float* __restrict__ out,
                           int Bn)
{
    __shared__ float featLDS[16 * FEAT_STRIDE]; // 50.4 KB: channel-major conv features, 16 samples
    __shared__ float partLDS[16 * 32];          // per-lane partial sums for mean
    __shared__ float mLDS[16 * 4];              // mean embeddings
    __shared__ float lseLDS[16];                // per-sample logsumexp
    __shared__ float logitsLDS[16 * 16];

    const int lane = threadIdx.x;   // wave32: one wave per block
    const int tile = blockIdx.x;    // 16 samples per tile

    // Uniform conv weights/bias (compiler scalarizes these)
    float cw[16];
    #pragma unroll
    for (int t = 0; t < 16; ++t) cw[t] = conv_w[t];
    float cb[4];
    #pragma unroll
    for (int t = 0; t < 4; ++t) cb[t] = conv_b[t];

    // ---- Phase 1: conv 2x2/s2 -> featLDS, fused mean partials ----
    // flat index k = c*196 + p, and k%4 == p%4 (196 % 4 == 0). Lane handles
    // p = lane, lane+32, ... so all its contributions go to mean dim d = lane%4.
    for (int s = 0; s < 16; ++s) {
        const int S = tile * 16 + s;
        float accm = 0.0f;
        if (S < Bn) {
            const float* xs = x + (size_t)S * 784;
            for (int p = lane; p < 196; p += 32) {
                const int i = p / 14;
                const int j = p - i * 14;
                const float* r0 = xs + (2 * i) * 28 + 2 * j;   // 8B aligned
                const v2f t0 = *(const v2f*)r0;
                const v2f t1 = *(const v2f*)(r0 + 28);
                float* dst = &featLDS[s * FEAT_STRIDE + p];
                #pragma unroll
                for (int c = 0; c < 4; ++c) {
                    float y = fmaf(t0.x, cw[4*c+0],
                              fmaf(t0.y, cw[4*c+1],
                              fmaf(t1.x, cw[4*c+2],
                              fmaf(t1.y, cw[4*c+3], cb[c]))));
                    dst[c * 196] = y;
                    accm += y;
                }
            }
        } else {
            for (int p = lane; p < 196; p += 32) {
                float* dst = &featLDS[s * FEAT_STRIDE + p];
                #pragma unroll
                for (int c = 0; c < 4; ++c) dst[c * 196] = 0.0f;
            }
        }
        partLDS[s * 32 + lane] = accm;
    }
    __syncthreads();

    // ---- Phase 2: mean embeddings m[s][d] = (1/196) * sum over p%4==d, all c ----
    for (int u = lane; u < 64; u += 32) {
        const int s = u & 15;
        const int d = u >> 4;        // lane pass 1: d=0/1, pass 2: d=2/3
        float sum = 0.0f;
        #pragma unroll
        for (int l = 0; l < 8; ++l) sum += partLDS[s * 32 + d + 4 * l];
        mLDS[s * 4 + d] = sum * (1.0f / 196.0f);
    }
    __syncthreads();

    // ---- Phase 3: logits = (feat .* m[k%4]) @ W^T via V_WMMA_F32_16X16X4_F32 ----
    // A (16x4): lanes 0-15 -> M=lane, K={k,k+1}; lanes 16-31 -> M=lane-16, K={k+2,k+3}
    // B (4x16): same lane split for K; N = lane&15 (classes 10..15 zeroed, branch-free)
    const int Mrow = lane & 15;
    const int koff = (lane >> 4) << 1;            // 0 or 2
    const float sc0 = mLDS[Mrow * 4 + koff];      // scale for K%4 == koff
    const float sc1 = mLDS[Mrow * 4 + koff + 1];  // scale for K%4 == koff+1
    const int Ncol = Mrow;
    const float bvalid = (Ncol < 10) ? 1.0f : 0.0f;
    const float* wrow = lin_w + (size_t)((Ncol < 10) ? Ncol : 0) * 784 + koff;
    const float* frow = &featLDS[Mrow * FEAT_STRIDE + koff];

    v8f acc = {};
    for (int k = 0; k < 784; k += 4) {
        v2f a = *(const v2f*)(frow + k);          // ds_load_b64, bank-conflict-free
        a.x *= sc0;
        a.y *= sc1;
        v2f b = *(const v2f*)(wrow + k);          // global_load_b64, hot in WGP$/L2
        b.x *= bvalid;
        b.y *= bvalid;
        acc = __builtin_amdgcn_wmma_f32_16x16x4_f32(
            /*neg_a=*/false, a, /*neg_b=*/false, b,
            /*c_mod=*/(short)0, acc, /*reuse_a=*/false, /*reuse_b=*/false);
    }

    // ---- Phase 4: bias + log_softmax over 10 classes ----
    // D layout: VGPR r holds M=r (lanes 0-15) / M=r+8 (lanes 16-31), N = lane&15
    const float bias = (Ncol < 10) ? lin_b[Ncol] : 0.0f;
    const int rbase = (lane >> 4) * 8;
    #pragma unroll
    for (int r = 0; r < 8; ++r)
        logitsLDS[(rbase + r) * 16 + Ncol] = acc[r] + bias;
    __syncthreads();

    if (lane < 16) {
        float mx = -3.402823466e38f;
        #pragma unroll
        for (int j = 0; j < 10; ++j) mx = fmaxf(mx, logitsLDS[lane * 16 + j]);
        float se = 0.0f;
        #pragma unroll
        for (int j = 0; j < 10; ++j) se += __expf(logitsLDS[lane * 16 + j] - mx);
        lseLDS[lane] = mx + __logf(se);
    }
    __syncthreads();

    for (int u = lane; u < 160; u += 32) {
        const int s = u / 10;
        const int S = tile * 16 + s;
        if (S < Bn)
            out[(size_t)tile * 160 + u] = logitsLDS[s * 16 + (u - s * 10)] - lseLDS[s];
    }
}

extern "C" void kernel_launch(void* const* d_in, const int* in_sizes, int n_in,
                              void* d_out, int out_size, void* d_ws, size_t ws_size,
                              hipStream_t stream) {
    const float* x      = (const float*)d_in[0];
    const float* conv_w = (const float*)d_in[1];
    const float* conv_b = (const float*)d_in[2];
    // d_in[3] (rotation_params) and d_in[4] (entangle_params) feed a softmax over a
    // length-1 axis -> exactly 1.0; they provably do not affect the output.
    const float* lin_w  = (const float*)d_in[5];
    const float* lin_b  = (const float*)d_in[6];
    float* out = (float*)d_out;

    const int Bn    = in_sizes[0] / 784;       // 32768
    const int tiles = (Bn + 15) / 16;          // one wave32 per 16-sample tile

    quanv_attn_cls_kernel<<<tiles, 32, 0, stream>>>(x, conv_w, conv_b, lin_w, lin_b, out, Bn);
}